// BottomSkip_88098369176171
// MI455X (gfx1250) — compile-verified
//
#include <hip/hip_runtime.h>
#include <math.h>

#define NN 50000
#define EE 400000
#define HIDW 256
#define NHEAD 4
#define DHEAD 64

typedef __attribute__((ext_vector_type(2))) float v2f;
typedef __attribute__((ext_vector_type(8))) float v8f;

// ---------------------------------------------------------------- utilities
__global__ void __launch_bounds__(256) fill_kernel(float* p, float v, int n) {
    for (int i = blockIdx.x * blockDim.x + threadIdx.x; i < n;
         i += gridDim.x * blockDim.x)
        p[i] = v;
}

__device__ __forceinline__ void atomicMaxF32(float* addr, float val) {
    if (val >= 0.0f)
        atomicMax((int*)addr, __float_as_int(val));
    else
        atomicMin((unsigned int*)addr, __float_as_uint(val));
}

// ---------------------------------------------------------------- WMMA GEMM
// C[M=NN, 256] = A[M, K] (row stride lda) @ W[K, 256]  (+bias) (+relu)
// Block = 128 threads = 4 waves; block -> 16 rows; wave -> 64 columns.
// FP32 WMMA 16x16x4: A frag v2f (lane<16: K=k,k+1 ; lane>=16: K=k+2,k+3),
// B frag v2f mirrored; D/C v8f per ISA layout (VGPR v: M = v + 8*(lane/16)).
__global__ void __launch_bounds__(128)
gemm_wmma_f32(const float* __restrict__ A, int lda,
              const float* __restrict__ W,
              const float* __restrict__ bias,
              float* __restrict__ C, int ldc,
              int K, int do_relu) {
    const int tileM = blockIdx.x * 16;
    const int wave  = threadIdx.x >> 5;
    const int lane  = threadIdx.x & 31;
    const int half  = lane >> 4;   // 0: lanes 0-15, 1: lanes 16-31
    const int l16   = lane & 15;
    const int tileN = wave * 64;

    v8f acc0 = {}, acc1 = {}, acc2 = {}, acc3 = {};
    const float* Arow = A + (long)(tileM + l16) * lda;

    for (int k = 0; k < K; k += 4) {
        v2f a;
        a.x = Arow[k + 2 * half];
        a.y = Arow[k + 2 * half + 1];
        const float* Wk = W + (long)(k + 2 * half) * HIDW + tileN + l16;
        v2f b0, b1, b2, b3;
        b0.x = Wk[0];  b0.y = Wk[HIDW];
        b1.x = Wk[16]; b1.y = Wk[16 + HIDW];
        b2.x = Wk[32]; b2.y = Wk[32 + HIDW];
        b3.x = Wk[48]; b3.y = Wk[48 + HIDW];
        acc0 = __builtin_amdgcn_wmma_f32_16x16x4_f32(false, a, false, b0,
                                                     (short)0, acc0, false, false);
        acc1 = __builtin_amdgcn_wmma_f32_16x16x4_f32(false, a, false, b1,
                                                     (short)0, acc1, false, false);
        acc2 = __builtin_amdgcn_wmma_f32_16x16x4_f32(false, a, false, b2,
                                                     (short)0, acc2, false, false);
        acc3 = __builtin_amdgcn_wmma_f32_16x16x4_f32(false, a, false, b3,
                                                     (short)0, acc3, false, false);
    }

    const int rowBase = tileM + 8 * half;
#pragma unroll
    for (int j = 0; j < 4; ++j) {
        v8f acc = (j == 0) ? acc0 : (j == 1) ? acc1 : (j == 2) ? acc2 : acc3;
        const int col = tileN + j * 16 + l16;
        const float bv = bias ? bias[col] : 0.0f;
#pragma unroll
        for (int v = 0; v < 8; ++v) {
            float val = acc[v] + bv;
            if (do_relu) val = fmaxf(val, 0.0f);
            C[(long)(rowBase + v) * ldc + col] = val;
        }
    }
}

// ---------------------------------------------------------------- GAT pieces
// el[n,h] = sum_d z[n,h*64+d]*al[h,d] ; er likewise
__global__ void __launch_bounds__(256)
attn_proj_kernel(const float* __restrict__ z, const float* __restrict__ al,
                 const float* __restrict__ ar, float* __restrict__ el,
                 float* __restrict__ er) {
    int i = blockIdx.x * blockDim.x + threadIdx.x;
    if (i >= NN * NHEAD) return;
    int n = i >> 2, h = i & 3;
    const float* zp = z + (long)n * HIDW + h * DHEAD;
    const float* alp = al + h * DHEAD;
    const float* arp = ar + h * DHEAD;
    float sl = 0.f, sr = 0.f;
#pragma unroll 4
    for (int d = 0; d < DHEAD; ++d) {
        float v = zp[d];
        sl += v * alp[d];
        sr += v * arp[d];
    }
    el[i] = sl;
    er[i] = sr;
}

__global__ void __launch_bounds__(256)
edge_attn_max_kernel(const int* __restrict__ src, const int* __restrict__ dst,
                     const float* __restrict__ el, const float* __restrict__ er,
                     float* __restrict__ e, float* __restrict__ m) {
    int i = blockIdx.x * blockDim.x + threadIdx.x;
    if (i >= EE * NHEAD) return;
    int ed = i >> 2, h = i & 3;
    int s = src[ed], d = dst[ed];
    float v = el[s * NHEAD + h] + er[d * NHEAD + h];
    v = (v > 0.f) ? v : 0.2f * v;   // leaky_relu(0.2)
    e[i] = v;
    atomicMaxF32(&m[d * NHEAD + h], v);
}

__global__ void __launch_bounds__(256)
edge_exp_sum_kernel(const int* __restrict__ dst, const float* __restrict__ e,
                    const float* __restrict__ m, float* __restrict__ ex,
                    float* __restrict__ den) {
    int i = blockIdx.x * blockDim.x + threadIdx.x;
    if (i >= EE * NHEAD) return;
    int ed = i >> 2, h = i & 3;
    int d = dst[ed];
    float v = expf(e[i] - m[d * NHEAD + h]);
    ex[i] = v;
    atomicAdd(&den[d * NHEAD + h], v);
}

// one block per edge, 256 threads = channels
__global__ void __launch_bounds__(256)
edge_scatter_kernel(const int* __restrict__ src, const int* __restrict__ dst,
                    const float* __restrict__ ex, const float* __restrict__ den,
                    const float* __restrict__ z, float* __restrict__ rst) {
    int ed = blockIdx.x;
    int c = threadIdx.x;
    __shared__ float a_sh[NHEAD];
    int d = dst[ed];
    if (c < NHEAD)
        a_sh[c] = ex[ed * NHEAD + c] / fmaxf(den[d * NHEAD + c], 1e-9f);
    __syncthreads();
    int s = src[ed];
    atomicAdd(&rst[(long)d * HIDW + c], a_sh[c >> 6] * z[(long)s * HIDW + c]);
}

__global__ void __launch_bounds__(256)
bias_relu_kernel(const float* __restrict__ rst, const float* __restrict__ b,
                 float* __restrict__ out, int ldout) {
    int i = blockIdx.x * blockDim.x + threadIdx.x;
    if (i >= NN * HIDW) return;
    int n = i >> 8, c = i & 255;
    out[(long)n * ldout + c] = fmaxf(rst[i] + b[c], 0.0f);
}

__global__ void __launch_bounds__(256)
copy_feats_kernel(const float* __restrict__ f, float* __restrict__ hcat) {
    int i = blockIdx.x * blockDim.x + threadIdx.x;
    if (i >= NN * 128) return;
    int n = i >> 7, c = i & 127;
    hcat[(long)n * 384 + HIDW + c] = f[i];
}

// ---------------------------------------------------------------- GIN pieces
__global__ void __launch_bounds__(256)
edge_agg_kernel(const int* __restrict__ src, const int* __restrict__ dst,
                const float* __restrict__ h, float* __restrict__ agg, int Kin) {
    int ed = blockIdx.x;
    int s = src[ed], d = dst[ed];
    for (int c = threadIdx.x; c < Kin; c += blockDim.x)
        atomicAdd(&agg[(long)d * Kin + c], h[(long)s * Kin + c]);
}

__global__ void __launch_bounds__(256)
build_x_kernel(const float* __restrict__ h, const float* __restrict__ agg,
               const float* __restrict__ eps_ptr, float* __restrict__ x, int n) {
    float e1 = 1.0f + eps_ptr[0];
    int i = blockIdx.x * blockDim.x + threadIdx.x;
    if (i < n) x[i] = e1 * h[i] + agg[i];
}

// two-stage column reduction: block-local partials then 1 atomic/column/block
__global__ void __launch_bounds__(256)
bn_stats_kernel(const float* __restrict__ x, float* __restrict__ sum,
                float* __restrict__ sumsq) {
    int c = threadIdx.x;
    float s = 0.f, sq = 0.f;
    for (int n = blockIdx.x; n < NN; n += gridDim.x) {
        float v = x[(long)n * HIDW + c];
        s += v;
        sq += v * v;
    }
    atomicAdd(&sum[c], s);
    atomicAdd(&sumsq[c], sq);
}

__global__ void __launch_bounds__(256)
bn_finalize_kernel(const float* __restrict__ s, const float* __restrict__ sq,
                   const float* __restrict__ g, const float* __restrict__ be,
                   float* __restrict__ sc, float* __restrict__ sh, float invn) {
    int c = threadIdx.x;
    float mu = s[c] * invn;
    float var = sq[c] * invn - mu * mu;
    float scale = g[c] * rsqrtf(var + 1e-5f);
    sc[c] = scale;
    sh[c] = be[c] - mu * scale;
}

__global__ void __launch_bounds__(256)
bn_apply_relu_kernel(float* __restrict__ x, const float* __restrict__ sc,
                     const float* __restrict__ sh, int n) {
    int i = blockIdx.x * blockDim.x + threadIdx.x;
    if (i >= n) return;
    int c = i & 255;
    x[i] = fmaxf(x[i] * sc[c] + sh[c], 0.0f);
}

// ---------------------------------------------------------------- host side
static void launch_fill(hipStream_t st, float* p, float v, int n) {
    int blocks = (n + 255) / 256;
    if (blocks > 4096) blocks = 4096;
    fill_kernel<<<blocks, 256, 0, st>>>(p, v, n);
}

static void run_gat(hipStream_t st, const float* hin, int K, const float* W,
                    const float* al, const float* ar, const float* b,
                    const int* src, const int* dst, float* z, float* rst,
                    float* e, float* ex, float* el, float* er, float* m,
                    float* den, float* out, int ldout) {
    gemm_wmma_f32<<<NN / 16, 128, 0, st>>>(hin, K, W, nullptr, z, HIDW, K, 0);
    attn_proj_kernel<<<(NN * NHEAD + 255) / 256, 256, 0, st>>>(z, al, ar, el, er);
    launch_fill(st, m, -INFINITY, NN * NHEAD);
    launch_fill(st, den, 0.0f, NN * NHEAD);
    launch_fill(st, rst, 0.0f, NN * HIDW);
    edge_attn_max_kernel<<<(EE * NHEAD + 255) / 256, 256, 0, st>>>(src, dst, el,
                                                                   er, e, m);
    edge_exp_sum_kernel<<<(EE * NHEAD + 255) / 256, 256, 0, st>>>(dst, e, m, ex,
                                                                  den);
    edge_scatter_kernel<<<EE, HIDW, 0, st>>>(src, dst, ex, den, z, rst);
    bias_relu_kernel<<<(NN * HIDW + 255) / 256, 256, 0, st>>>(rst, b, out, ldout);
}

static void run_gin(hipStream_t st, const float* hin, int Kin, const float* eps,
                    const float* W1, const float* b1, const float* g1,
                    const float* be1, const float* W2, const float* b2,
                    const int* src, const int* dst, float* agg, float* xbuf,
                    float* ybuf, float* colsum, float* colsq, float* bnsc,
                    float* bnsh, float* out, int ldc_out) {
    launch_fill(st, agg, 0.0f, NN * Kin);
    edge_agg_kernel<<<EE, 256, 0, st>>>(src, dst, hin, agg, Kin);
    build_x_kernel<<<(NN * Kin + 255) / 256, 256, 0, st>>>(hin, agg, eps, xbuf,
                                                           NN * Kin);
    gemm_wmma_f32<<<NN / 16, 128, 0, st>>>(xbuf, Kin, W1, b1, ybuf, HIDW, Kin, 0);
    launch_fill(st, colsum, 0.0f, HIDW);
    launch_fill(st, colsq, 0.0f, HIDW);
    bn_stats_kernel<<<256, 256, 0, st>>>(ybuf, colsum, colsq);
    bn_finalize_kernel<<<1, 256, 0, st>>>(colsum, colsq, g1, be1, bnsc, bnsh,
                                          1.0f / (float)NN);
    bn_apply_relu_kernel<<<(NN * HIDW + 255) / 256, 256, 0, st>>>(ybuf, bnsc,
                                                                  bnsh, NN * HIDW);
    gemm_wmma_f32<<<NN / 16, 128, 0, st>>>(ybuf, HIDW, W2, b2, out, ldc_out,
                                           HIDW, 1);
}

extern "C" void kernel_launch(void* const* d_in, const int* in_sizes, int n_in,
                              void* d_out, int out_size, void* d_ws,
                              size_t ws_size, hipStream_t stream) {
    (void)in_sizes; (void)n_in; (void)out_size; (void)ws_size;
    const float* feats   = (const float*)d_in[0];
    const float* gat0_W  = (const float*)d_in[5];
    const float* gat0_al = (const float*)d_in[6];
    const float* gat0_ar = (const float*)d_in[7];
    const float* gat0_b  = (const float*)d_in[8];
    const float* gat1_W  = (const float*)d_in[9];
    const float* gat1_al = (const float*)d_in[10];
    const float* gat1_ar = (const float*)d_in[11];
    const float* gat1_b  = (const float*)d_in[12];
    const float* gin0_eps = (const float*)d_in[13];
    const float* gin0_W1  = (const float*)d_in[14];
    const float* gin0_b1  = (const float*)d_in[15];
    const float* gin0_g1  = (const float*)d_in[16];
    const float* gin0_be1 = (const float*)d_in[17];
    const float* gin0_W2  = (const float*)d_in[18];
    const float* gin0_b2  = (const float*)d_in[19];
    const float* gin1_eps = (const float*)d_in[20];
    const float* gin1_W1  = (const float*)d_in[21];
    const float* gin1_b1  = (const float*)d_in[22];
    const float* gin1_g1  = (const float*)d_in[23];
    const float* gin1_be1 = (const float*)d_in[24];
    const float* gin1_W2  = (const float*)d_in[25];
    const float* gin1_b2  = (const float*)d_in[26];
    float* out = (float*)d_out;   // [NN, 512]

    // workspace layout (floats)
    float* ws = (float*)d_ws;
    float* bufA = ws;                         // N*384
    float* bufB = bufA + (size_t)NN * 384;    // N*384
    float* bufC = bufB + (size_t)NN * 384;    // N*384
    float* e_buf  = bufC + (size_t)NN * 384;  // E*4
    float* ex_buf = e_buf + (size_t)EE * 4;   // E*4
    float* el  = ex_buf + (size_t)EE * 4;     // N*4
    float* er  = el + (size_t)NN * 4;         // N*4
    float* m   = er + (size_t)NN * 4;         // N*4
    float* den = m + (size_t)NN * 4;          // N*4
    float* colsum = den + (size_t)NN * 4;     // 256
    float* colsq  = colsum + 256;             // 256
    float* bnsc   = colsq + 256;              // 256
    float* bnsh   = bnsc + 256;               // 256

    for (int t = 0; t < 2; ++t) {
        const int* src = (const int*)d_in[t == 0 ? 1 : 3];
        const int* dst = (const int*)d_in[t == 0 ? 2 : 4];

        // GAT layer 0: feats[N,128] -> bufA[N,256]
        run_gat(stream, feats, 128, gat0_W + (size_t)t * 128 * 256,
                gat0_al + t * 256, gat0_ar + t * 256, gat0_b + t * 256, src, dst,
                bufB, bufC, e_buf, ex_buf, el, er, m, den, bufA, 256);
        // GAT layer 1: bufA[N,256] -> bufA with row stride 384 (cols 0..255)
        run_gat(stream, bufA, 256, gat1_W + (size_t)t * 256 * 256,
                gat1_al + t * 256, gat1_ar + t * 256, gat1_b + t * 256, src, dst,
                bufB, bufC, e_buf, ex_buf, el, er, m, den, bufA, 384);
        // skip-concat: bufA[:,256:384] = feats
        copy_feats_kernel<<<(NN * 128 + 255) / 256, 256, 0, stream>>>(feats, bufA);

        // GIN layer 0: hin=bufA(384), agg=bufC, x=bufB, y=bufA, out=bufB(256)
        run_gin(stream, bufA, 384, gin0_eps + t, gin0_W1 + (size_t)t * 384 * 256,
                gin0_b1 + t * 256, gin0_g1 + t * 256, gin0_be1 + t * 256,
                gin0_W2 + (size_t)t * 256 * 256, gin0_b2 + t * 256, src, dst,
                bufC, bufB, bufA, colsum, colsq, bnsc, bnsh, bufB, 256);
        // GIN layer 1: hin=bufB(256), agg=bufC, x=bufA, y=bufC,
        //              out -> d_out columns [t*256, t*256+256) with ldc=512
        run_gin(stream, bufB, 256, gin1_eps + t, gin1_W1 + (size_t)t * 256 * 256,
                gin1_b1 + t * 256, gin1_g1 + t * 256, gin1_be1 + t * 256,
                gin1_W2 + (size_t)t * 256 * 256, gin1_b2 + t * 256, src, dst,
                bufC, bufA, bufC, colsum, colsq, bnsc, bnsh, out + t * 256, 512);
    }
}